// QCModel_68882685493537
// MI455X (gfx1250) — compile-verified
//
#include <hip/hip_runtime.h>
#include <hip/hip_bf16.h>
#include <stdint.h>

// scores[i,j] = -sum_k relu(q[i,k] - c[j,k]),  Q=2048, C=8192, D=16, all f32.
//
// Roofline (MI455X): 67 MB of output stores -> ~2.9 us floor @ 23.3 TB/s.
// Compute: ~47 VALU ops/output (16 sub, 16 max, 15 add) dominates (~3x floor),
// so the kernel minimizes the inner body, keeps c-rows in VGPRs, reads q-rows
// through the uniform/scalar path, and uses the CDNA5 async global->LDS engine
// (GLOBAL_LOAD_ASYNC_TO_LDS_B128 + s_wait_asynccnt) for the corpus tile stage.
// WMMA does not apply: relu(q-c) is not bilinear, and f32 WMMA is 16x16x4 only.

#define QN 2048
#define CN 8192
#define DN 16
#define TJ 256   // corpus rows per block (1 per thread)
#define TI 32    // query rows per block

__global__ __launch_bounds__(TJ) void qc_relu_l1_kernel(
    const float* __restrict__ q,    // [QN, DN]
    const float* __restrict__ c,    // [CN, DN]
    float* __restrict__ out) {      // [QN, CN]
  __shared__ float ctile[TJ * DN];  // 16 KB staging for this block's c rows

  const int t  = threadIdx.x;
  const int j  = blockIdx.x * TJ + t;      // this thread's corpus row
  const int i0 = blockIdx.y * TI;          // first query row of this block

  // ---- CDNA5 async-tensor path: stage c row j (64 B) into LDS -------------
  // INST_OFFSET is added to both the global and the LDS address, so four
  // b128 ops with offset 0/16/32/48 copy the row verbatim. Tracked by
  // ASYNCcnt; each thread only reads back its own row, so a per-wave
  // s_wait_asynccnt 0 is the full dependency.
  {
    unsigned long long ga = (unsigned long long)(uintptr_t)(c + (size_t)j * DN);
    unsigned lds = (unsigned)(uintptr_t)
        (__attribute__((address_space(3))) void*)&ctile[t * DN];
    asm volatile("global_load_async_to_lds_b128 %0, %1, off"
                 :: "v"(lds), "v"(ga) : "memory");
    asm volatile("global_load_async_to_lds_b128 %0, %1, off offset:16"
                 :: "v"(lds), "v"(ga) : "memory");
    asm volatile("global_load_async_to_lds_b128 %0, %1, off offset:32"
                 :: "v"(lds), "v"(ga) : "memory");
    asm volatile("global_load_async_to_lds_b128 %0, %1, off offset:48"
                 :: "v"(lds), "v"(ga) : "memory");
    asm volatile("s_wait_asynccnt 0" ::: "memory");
  }
  __syncthreads();

  // c row -> 16 VGPRs (4x ds_load_b128)
  float cr[DN];
#pragma unroll
  for (int u = 0; u < 4; ++u) {
    float4 v = *(const float4*)&ctile[t * DN + 4 * u];
    cr[4 * u + 0] = v.x; cr[4 * u + 1] = v.y;
    cr[4 * u + 2] = v.z; cr[4 * u + 3] = v.w;
  }

  // ---- main loop: one query row per iteration -----------------------------
  // q address is uniform across the block per iteration -> scalar (SMEM) loads.
#pragma unroll 2
  for (int ii = 0; ii < TI; ++ii) {
    const float* __restrict__ qr = q + (size_t)(i0 + ii) * DN;

    float r[DN];
#pragma unroll
    for (int k = 0; k < DN; ++k) {
      float d = qr[k] - cr[k];        // v_sub with scalar operand
      r[k] = fmaxf(d, 0.0f);          // v_max with inline 0
    }
    // log-depth pairwise tree: 15 adds, depth 4 -> good ILP
#pragma unroll
    for (int s = 8; s >= 1; s >>= 1)
#pragma unroll
      for (int k = 0; k < s; ++k) r[k] += r[k + s];

    // lanes t=0..255 -> consecutive j -> fully coalesced; output is
    // write-once, never re-read in-kernel -> non-temporal store.
    __builtin_nontemporal_store(-r[0], &out[(size_t)(i0 + ii) * CN + j]);
  }
}

extern "C" void kernel_launch(void* const* d_in, const int* in_sizes, int n_in,
                              void* d_out, int out_size, void* d_ws, size_t ws_size,
                              hipStream_t stream) {
  (void)in_sizes; (void)n_in; (void)d_ws; (void)ws_size; (void)out_size;
  const float* q = (const float*)d_in[0];   // [2048, 16] f32
  const float* c = (const float*)d_in[1];   // [8192, 16] f32
  float* out = (float*)d_out;               // [2048, 8192] f32

  dim3 grid(CN / TJ, QN / TI);  // 32 x 64 = 2048 blocks
  dim3 block(TJ);               // 256 threads = 8 wave32
  qc_relu_l1_kernel<<<grid, block, 0, stream>>>(q, c, out);
}